// SWD11_28449863369555
// MI455X (gfx1250) — compile-verified
//
#include <hip/hip_runtime.h>
#include <hip/hip_bf16.h>
#include <stdint.h>

// ---------------------------------------------------------------------------
// SWD11: per-column stable sort over L + neighbor-column rank gather.
//   out[bh,i,d] = sum_{s=1..3} sortedv[bh,d][ rank[bh,(d+s)&63][i] ]
// B*H = 64, L = 4096, D = 64  ->  4096 independent columns of 4096 fp32.
//
// Kernel 1 (sort): one workgroup per column. TDM (tensor_load_to_lds) DMAs
//   the stride-256B column into LDS (overlapped with payload init); bitonic
//   sort on composite key (value, original index) == stable argsort. Stages
//   with j<=4 (and the whole k=2,4,8 prefix) run register-resident on 8
//   contiguous elements per thread: 55 LDS round-trips instead of 78.
// Kernel 2 (gather): L2-resident gather (per-bh working set 2MB << 192MB L2).
//
// Scratch per bh = 64 cols * 4096 * (4B val + 4B rank) = 2 MB. kernel_launch
// chunks over bh based on the real ws_size (1 chunk when ws >= 128MB).
// ---------------------------------------------------------------------------

typedef __attribute__((ext_vector_type(4))) unsigned int u32x4;
typedef __attribute__((ext_vector_type(8))) int          i32x8;
typedef __attribute__((ext_vector_type(4))) int          i32x4;

#if __has_include(<hip/amd_detail/amd_gfx1250_TDM.h>)
#define TDM_ARITY6 1   // amdgpu-toolchain / therock headers -> 6-arg builtin
#else
#define TDM_ARITY6 0   // ROCm 7.2 -> 5-arg builtin
#endif

static constexpr int LDIM = 4096;               // sort axis length
static constexpr int DDIM = 64;                 // head dim
static constexpr int NBH  = 64;                 // B*H
static constexpr int NTOT = NBH * DDIM * LDIM;  // 16,777,216 per output copy

__device__ __forceinline__
void cmpswap(float& a, float& b, int& ia, int& ib, bool up)
{
    const bool gt = (a > b) || ((a == b) && (ia > ib));
    if (gt == up) {
        float tv = a; a = b; b = tv;
        int   ti = ia; ia = ib; ib = ti;
    }
}

__global__ __launch_bounds__(512)
void swd11_sort_columns(const float* __restrict__ v,
                        float* __restrict__ sortedv,
                        int* __restrict__ rank,
                        int col0)
{
    __shared__ __align__(16) float vals[LDIM];   // 16 KB
    __shared__ __align__(16) int   idxs[LDIM];   // 16 KB
    const int lc  = blockIdx.x;          // local column index (scratch slot)
    const int col = col0 + lc;           // global column = bh*64 + d
    const int tid = threadIdx.x;
    const float* gcol = v + ((size_t)(col >> 6) * (size_t)LDIM * DDIM
                             + (size_t)(col & 63));

    // payload init overlaps with the TDM transfer
    for (int i = tid; i < LDIM; i += 512) idxs[i] = i;

#if defined(__gfx1250__)
    if (tid < 32) {  // wave 0 only issues the DMA (TDM ignores EXEC per-lane)
        const uint32_t lds_off = (uint32_t)(uintptr_t)&vals[0];
        const uint64_t ga      = (uint64_t)(uintptr_t)gcol;
        // D# group 0: count=1 | lds_addr | global_addr[56:0] | type=2
        u32x4 g0 = { 1u,
                     lds_off,
                     (uint32_t)ga,
                     ((uint32_t)(ga >> 32) & 0x01FFFFFFu) | (2u << 30) };
        // D# group 1: data_size=2 (4B); tensor_dim0=64, tensor_dim1=4096;
        //             tile_dim0=1, tile_dim1=4096; tensor_dim0_stride=64 elems
        i32x8 g1 = { (int)0x00020000u,        // workgroup_mask=0, data_size=4B
                     (int)(64u   << 16),      // tensor_dim0 lo16 = 64
                     (int)(4096u << 16),      // dim0 hi=0 | tensor_dim1 lo16 = 4096
                     (int)(1u    << 16),      // dim1 hi=0 | tile_dim0 = 1
                     (int)4096,               // tile_dim1 = 4096, tile_dim2 = 0
                     (int)64,                 // tensor_dim0_stride lo32 = 64
                     0, 0 };                  // stride hi / tensor_dim1_stride = 0
        // groups 2/3: benign higher-dim extents (=1) so nothing clips
        i32x4 g2 = { 1, 1, 0, 0 };               // tensor_dim2=1, tensor_dim3=1
        i32x4 g3 = { 0, (int)(1u << 16), 0, 0 }; // tensor_dim4=1, tile_dim4=0
#if TDM_ARITY6
        i32x8 gx = { 0, 0, 0, 0, 0, 0, 0, 0 };
        __builtin_amdgcn_tensor_load_to_lds(g0, g1, g2, g3, gx, 0);
#else
        __builtin_amdgcn_tensor_load_to_lds(g0, g1, g2, g3, 0);
#endif
        __builtin_amdgcn_s_wait_tensorcnt(0);
    }
#else
    // host-pass / non-gfx1250 parse path: plain strided loads
    for (int i = tid; i < LDIM; i += 512) vals[i] = gcol[(size_t)i * DDIM];
#endif
    __syncthreads();

    // ---- Bitonic sort, ascending on unique composite key (val, idx) ----
    // 512 threads * 8 contiguous elements: stages with j<=4 are register-
    // resident (one LDS round-trip for 3-6 network passes).

    // Prefix: k = 2, 4, 8 fully in registers.
    {
        const int base = tid << 3;
        float e[8]; int ix[8];
        #pragma unroll
        for (int r = 0; r < 8; ++r) { e[r] = vals[base + r]; ix[r] = idxs[base + r]; }
        // k=2 (j=1): dir = ((i&2)==0)
        cmpswap(e[0], e[1], ix[0], ix[1], true);
        cmpswap(e[2], e[3], ix[2], ix[3], false);
        cmpswap(e[4], e[5], ix[4], ix[5], true);
        cmpswap(e[6], e[7], ix[6], ix[7], false);
        // k=4 (j=2,1): dir = ((i&4)==0)
        cmpswap(e[0], e[2], ix[0], ix[2], true);
        cmpswap(e[1], e[3], ix[1], ix[3], true);
        cmpswap(e[4], e[6], ix[4], ix[6], false);
        cmpswap(e[5], e[7], ix[5], ix[7], false);
        cmpswap(e[0], e[1], ix[0], ix[1], true);
        cmpswap(e[2], e[3], ix[2], ix[3], true);
        cmpswap(e[4], e[5], ix[4], ix[5], false);
        cmpswap(e[6], e[7], ix[6], ix[7], false);
        // k=8 (j=4,2,1): dir constant per 8-block
        const bool u8 = ((base & 8) == 0);
        cmpswap(e[0], e[4], ix[0], ix[4], u8);
        cmpswap(e[1], e[5], ix[1], ix[5], u8);
        cmpswap(e[2], e[6], ix[2], ix[6], u8);
        cmpswap(e[3], e[7], ix[3], ix[7], u8);
        cmpswap(e[0], e[2], ix[0], ix[2], u8);
        cmpswap(e[1], e[3], ix[1], ix[3], u8);
        cmpswap(e[4], e[6], ix[4], ix[6], u8);
        cmpswap(e[5], e[7], ix[5], ix[7], u8);
        #pragma unroll
        for (int r = 0; r < 8; r += 2) cmpswap(e[r], e[r+1], ix[r], ix[r+1], u8);
        #pragma unroll
        for (int r = 0; r < 8; ++r) { vals[base + r] = e[r]; idxs[base + r] = ix[r]; }
    }
    __syncthreads();

    for (int k = 16; k <= LDIM; k <<= 1) {
        // LDS passes: j = k/2 .. 8
        for (int j = k >> 1; j >= 8; j >>= 1) {
            #pragma unroll 4
            for (int p = tid; p < LDIM / 2; p += 512) {
                const int i = (p << 1) - (p & (j - 1)); // bit j of i is 0
                const int m = i | j;                    // partner = i + j
                float a = vals[i], b = vals[m];
                int  ia = idxs[i], ib = idxs[m];
                const bool gt = (a > b) || ((a == b) && (ia > ib));
                const bool up = ((i & k) == 0);
                if (gt == up) {
                    vals[i] = b; vals[m] = a;
                    idxs[i] = ib; idxs[m] = ia;
                }
            }
            __syncthreads();
        }
        // Register pass: j = 4, 2, 1 (direction constant across the 8-block)
        {
            const int base = tid << 3;
            const bool up = ((base & k) == 0);
            float e[8]; int ix[8];
            #pragma unroll
            for (int r = 0; r < 8; ++r) { e[r] = vals[base + r]; ix[r] = idxs[base + r]; }
            cmpswap(e[0], e[4], ix[0], ix[4], up);
            cmpswap(e[1], e[5], ix[1], ix[5], up);
            cmpswap(e[2], e[6], ix[2], ix[6], up);
            cmpswap(e[3], e[7], ix[3], ix[7], up);
            cmpswap(e[0], e[2], ix[0], ix[2], up);
            cmpswap(e[1], e[3], ix[1], ix[3], up);
            cmpswap(e[4], e[6], ix[4], ix[6], up);
            cmpswap(e[5], e[7], ix[5], ix[7], up);
            #pragma unroll
            for (int r = 0; r < 8; r += 2) cmpswap(e[r], e[r+1], ix[r], ix[r+1], up);
            #pragma unroll
            for (int r = 0; r < 8; ++r) { vals[base + r] = e[r]; idxs[base + r] = ix[r]; }
        }
        __syncthreads();
    }

    float* sv = sortedv + (size_t)lc * LDIM;
    int*   rk = rank    + (size_t)lc * LDIM;
    for (int i = tid; i < LDIM; i += 512) {
        sv[i] = vals[i];          // coalesced
        rk[idxs[i]] = i;          // scatter: rank of original row idxs[i]
    }
}

__global__ __launch_bounds__(256)
void swd11_gather(const float* __restrict__ sortedv,
                  const int* __restrict__ rank,
                  float* __restrict__ out,
                  float* __restrict__ out2,
                  int bh0)
{
    const int t   = blockIdx.x * 256 + threadIdx.x;
    const int i   = t & (LDIM - 1);
    const int d   = (t >> 12) & 63;
    const int bhl = t >> 18;            // local bh within this chunk
    const int bh  = bh0 + bhl;          // global bh

    const float* sv  = sortedv + ((size_t)(bhl * 64 + d) << 12);
    const int*   rkb = rank    + ((size_t)bhl << 18);   // bhl * 64 * 4096

    float acc = 0.f;
    #pragma unroll
    for (int s = 1; s <= 3; ++s) {
        const int dp = (d + s) & 63;
        const int r  = rkb[((size_t)dp << 12) + i];     // coalesced over i
        acc += sv[r];                                   // random, L2-resident
    }
    const size_t o = (((size_t)bh << 12) + (size_t)i) * 64 + (size_t)d;
    out[o]  = acc;
    out2[o] = acc;   // reference returns (out, out)
}

extern "C" void kernel_launch(void* const* d_in, const int* in_sizes, int n_in,
                              void* d_out, int out_size, void* d_ws, size_t ws_size,
                              hipStream_t stream)
{
    // inputs: q, k, v (q/k unused by the math)
    const float* v = (const float*)d_in[2];

    // 2 MB of scratch per bh slice (64 columns * 4096 * 8B). Chunk by bh so
    // any ws_size >= 2MB works; ws >= 128MB -> single pass over all 64 bh.
    const size_t perBH = (size_t)DDIM * LDIM * (sizeof(float) + sizeof(int));
    int chunk = (int)(ws_size / perBH);
    if (chunk > NBH) chunk = NBH;
    if (chunk < 1)   chunk = 1;

    float* sortedv = (float*)d_ws;
    int*   rank    = (int*)((char*)d_ws
                            + (size_t)chunk * DDIM * LDIM * sizeof(float));

    for (int bh0 = 0; bh0 < NBH; bh0 += chunk) {
        const int nbh = (NBH - bh0 < chunk) ? (NBH - bh0) : chunk;
        swd11_sort_columns<<<nbh * DDIM, 512, 0, stream>>>(v, sortedv, rank,
                                                           bh0 * DDIM);
        swd11_gather<<<nbh * 1024, 256, 0, stream>>>(sortedv, rank,
                                                     (float*)d_out,
                                                     (float*)d_out + (size_t)NTOT,
                                                     bh0);
    }
}